// RetrievalEnhancedMBO_26766236188953
// MI455X (gfx1250) — compile-verified
//
#include <hip/hip_runtime.h>
#include <float.h>

typedef __attribute__((ext_vector_type(2))) float v2f;
typedef __attribute__((ext_vector_type(8))) float v8f;

#define B_Q    128     // number of queries
#define D_DIM  64      // feature dim
#define K_RET  10      // retrieval set size
#define NBLK_MAX 512   // max partial-knn blocks (ws = NBLK*128*20*8 bytes)

__device__ __forceinline__ bool lex_less(float s1, int i1, float s2, int i2) {
    // strict (dist, index) lexicographic order — matches jax.lax.top_k tie-break
    return (s1 < s2) || (s1 == s2 && i1 < i2);
}

// ---------------------------------------------------------------------------
// Kernel 1: partial k-NN. Each block owns a contiguous chunk of pool columns.
// 8 waves cover the 128 queries (16 per wave via one WMMA C-tile). Each lane
// keeps a register-resident sorted top-10 for exactly one query (half of its
// 16 columns per step), candidates written to workspace.
// ---------------------------------------------------------------------------
__global__ __launch_bounds__(256) void knn_partial_kernel(
    const float* __restrict__ x, const float* __restrict__ pool_x, int N,
    float* __restrict__ cand_s, int* __restrict__ cand_i)
{
    __shared__ float sbuf[8 * 256];          // per-wave 16x16 score tile (1 KB each)

    const int tid  = threadIdx.x;
    const int wave = tid >> 5;
    const int lane = tid & 31;
    const int colc = lane & 15;              // column (N-dim) this lane loads / owns in C
    const int hi   = lane >> 4;              // lane half: rows/K upper half selector
    const int koff = hi << 1;                // K offset within 4-wide fragment {0,2}
    const int qloc = lane >> 1;              // query (0..15) this lane selects for
    const int q    = (wave << 4) + qloc;     // global query id
    const int h    = lane & 1;               // which 8-column half of the tile
    float* sw = sbuf + wave * 256;

    // ---- persistent A tile: 16 queries x K=64 as 16 fragments of 16x4 f32.
    // ISA layout: V0 holds K = 4i+koff, V1 holds K = 4i+koff+1, M = lane&15.
    const int m = (wave << 4) + colc;
    v2f a[16];
    const float* xp = x + m * D_DIM + koff;
#pragma unroll
    for (int i = 0; i < 16; ++i) a[i] = *(const v2f*)(xp + 4 * i);

    // ---- per-lane sorted top-k (ascending (s, idx))
    float best[K_RET]; int bidx[K_RET];
#pragma unroll
    for (int i = 0; i < K_RET; ++i) { best[i] = INFINITY; bidx[i] = 0x7fffffff; }

    // column chunk of this block (16-aligned)
    const int nblk = gridDim.x;
    const int cpb  = (((N + nblk - 1) / nblk) + 15) & ~15;
    const int j0   = blockIdx.x * cpb;

    for (int j = j0; j < j0 + cpb; j += 16) {
        if (j >= N) break;

        // B fragment: column j+colc, same K striping as A.
        int row = j + colc; if (row >= N) row = N - 1;
        const float* bp = pool_x + (size_t)row * D_DIM + koff;

        int prow = j + 16 + colc; if (prow >= N) prow = N - 1;
        __builtin_prefetch(pool_x + (size_t)prow * D_DIM, 0, 1);

        v2f bt[16];
        float ps = 0.0f;                     // ||pool row||^2 (half, per lane)
#pragma unroll
        for (int i = 0; i < 16; ++i) {
            bt[i] = *(const v2f*)(bp + 4 * i);
            ps += bt[i].x * bt[i].x + bt[i].y * bt[i].y;
        }
        ps += __shfl_xor(ps, 16, 32);        // combine lane halves -> full ||p||^2

        // 16 chained f32 WMMAs (two accumulators to relax the D->C chain)
        v8f c0 = {}; v8f c1 = {};
#pragma unroll
        for (int i = 0; i < 16; i += 2) {
            c0 = __builtin_amdgcn_wmma_f32_16x16x4_f32(
                     false, a[i],     false, bt[i],     (short)0, c0, false, false);
            c1 = __builtin_amdgcn_wmma_f32_16x16x4_f32(
                     false, a[i + 1], false, bt[i + 1], (short)0, c1, false, false);
        }

        // ranking key s = ||p||^2 - 2 * <x, p>  (monotone in the true distance)
        // C layout: VGPR r -> (M = r + 8*hi, N = colc)
#pragma unroll
        for (int r = 0; r < 8; ++r) {
            float s = ps - 2.0f * (c0[r] + c1[r]);
            sw[(r + (hi << 3)) * 16 + colc] = s;
        }

        // redistribute: lane owns query qloc, columns h*8..h*8+7 (same-wave LDS
        // ops are in-order; per-wave buffers are disjoint -> no barrier needed)
#pragma unroll
        for (int t = 0; t < 8; ++t) {
            const int n   = (h << 3) + t;
            const float s = sw[(qloc << 4) + n];
            const int idx = j + n;
            if (idx < N && lex_less(s, idx, best[K_RET - 1], bidx[K_RET - 1])) {
                float cs = s; int ci = idx;
#pragma unroll
                for (int i = 0; i < K_RET; ++i) {
                    const bool take = lex_less(cs, ci, best[i], bidx[i]);
                    const float ts = take ? best[i] : cs;
                    const int   ti = take ? bidx[i] : ci;
                    best[i] = take ? cs : best[i];
                    bidx[i] = take ? ci : bidx[i];
                    cs = ts; ci = ti;
                }
            }
        }
    }

    // candidates laid out per-query-contiguous for the merge pass
    const int base = (q * nblk + blockIdx.x) * (2 * K_RET) + h * K_RET;
#pragma unroll
    for (int i = 0; i < K_RET; ++i) {
        cand_s[base + i] = best[i];
        cand_i[base + i] = bidx[i];
    }
}

// ---------------------------------------------------------------------------
// Kernel 2: per-query merge of nblk*20 candidates -> ascending top-10, then
// gather pool_x rows + pool_y into the output [128, 10, 65].
// ---------------------------------------------------------------------------
__global__ __launch_bounds__(256) void knn_merge_gather_kernel(
    const float* __restrict__ cand_s, const int* __restrict__ cand_i,
    const float* __restrict__ pool_x, const float* __restrict__ pool_y,
    int N, int nblk, float* __restrict__ out)
{
    const int q   = blockIdx.x;
    const int tid = threadIdx.x;

    __shared__ float ls[256 * K_RET];
    __shared__ int   li[256 * K_RET];
    __shared__ float rs[256];
    __shared__ int   ri[256];
    __shared__ int   rpp[256];
    __shared__ int   outi[K_RET];

    // phase 1: thread-local top-10 over a strided slice of the candidates
    float best[K_RET]; int bidx[K_RET];
#pragma unroll
    for (int i = 0; i < K_RET; ++i) { best[i] = INFINITY; bidx[i] = 0x7fffffff; }

    const int M    = nblk * 2 * K_RET;
    const int base = q * M;
    for (int e = tid; e < M; e += 256) {
        const float s = cand_s[base + e];
        const int idx = cand_i[base + e];
        if (lex_less(s, idx, best[K_RET - 1], bidx[K_RET - 1])) {
            float cs = s; int ci = idx;
#pragma unroll
            for (int i = 0; i < K_RET; ++i) {
                const bool take = lex_less(cs, ci, best[i], bidx[i]);
                const float ts = take ? best[i] : cs;
                const int   ti = take ? bidx[i] : ci;
                best[i] = take ? cs : best[i];
                bidx[i] = take ? ci : bidx[i];
                cs = ts; ci = ti;
            }
        }
    }
#pragma unroll
    for (int i = 0; i < K_RET; ++i) {
        ls[tid * K_RET + i] = best[i];
        li[tid * K_RET + i] = bidx[i];
    }
    __syncthreads();

    // phase 2: 10 rounds of global argmin (ascending order, lex tie-break)
    for (int r = 0; r < K_RET; ++r) {
        float bs = INFINITY; int bi = 0x7fffffff; int bp = -1;
#pragma unroll
        for (int i = 0; i < K_RET; ++i) {
            const int p = tid * K_RET + i;
            if (lex_less(ls[p], li[p], bs, bi)) { bs = ls[p]; bi = li[p]; bp = p; }
        }
        rs[tid] = bs; ri[tid] = bi; rpp[tid] = bp;
        __syncthreads();
        for (int off = 128; off > 0; off >>= 1) {
            if (tid < off) {
                if (lex_less(rs[tid + off], ri[tid + off], rs[tid], ri[tid])) {
                    rs[tid] = rs[tid + off];
                    ri[tid] = ri[tid + off];
                    rpp[tid] = rpp[tid + off];
                }
            }
            __syncthreads();
        }
        if (tid == 0) {
            outi[r] = ri[0];
            const int p = rpp[0];
            if (p >= 0) { ls[p] = INFINITY; li[p] = 0x7fffffff; }
        }
        __syncthreads();
    }

    // phase 3: gather [10, 65] rows for this query
    for (int t = tid; t < K_RET * (D_DIM + 1); t += 256) {
        const int e = t / (D_DIM + 1);
        const int c = t % (D_DIM + 1);
        int idx = outi[e];
        if (idx < 0 || idx >= N) idx = 0;    // safety only
        const float v = (c < D_DIM) ? pool_x[(size_t)idx * D_DIM + c] : pool_y[idx];
        out[((size_t)q * K_RET + e) * (D_DIM + 1) + c] = v;
    }
}

// ---------------------------------------------------------------------------
extern "C" void kernel_launch(void* const* d_in, const int* in_sizes, int n_in,
                              void* d_out, int out_size, void* d_ws, size_t ws_size,
                              hipStream_t stream) {
    const float* x      = (const float*)d_in[0];
    const float* pool_x = (const float*)d_in[1];
    const float* pool_y = (const float*)d_in[2];
    const int N = in_sizes[1] / D_DIM;       // 500000

    // workspace: cand_s + cand_i, each nblk*128*20 elements; size nblk to fit.
    const size_t per_blk = (size_t)B_Q * 2 * K_RET * (sizeof(float) + sizeof(int));
    int nblk = (int)(ws_size / per_blk);
    if (nblk > NBLK_MAX) nblk = NBLK_MAX;
    if (nblk < 1) nblk = 1;

    float* cand_s = (float*)d_ws;
    int*   cand_i = (int*)((char*)d_ws + sizeof(float) * (size_t)nblk * B_Q * 2 * K_RET);

    knn_partial_kernel<<<dim3(nblk), dim3(256), 0, stream>>>(
        x, pool_x, N, cand_s, cand_i);
    knn_merge_gather_kernel<<<dim3(B_Q), dim3(256), 0, stream>>>(
        cand_s, cand_i, pool_x, pool_y, N, nblk, (float*)d_out);
}